// LowRankRedistributor_38689065402635
// MI455X (gfx1250) — compile-verified
//
#include <hip/hip_runtime.h>
#include <math.h>

// ---------------------------------------------------------------------------
// LowRankRedistributor for MI455X (gfx1250, wave32, WMMA f32 16x16x4)
//
//   out = c1 * M0 + c2 * logits,  M0 = logits @ W @ W^T
// with c1, c2 computed from ||logits||, ||M0||, <M0, logits> (all norms fold
// into scalars because everything before the final norm is linear).
//
// WMMA trick: D rows are independent; rows 8..15 of the 16-row tile are
// discarded, so A rows 8..15 are fed duplicated data (m & 7) instead of
// zeros -> no exec-mask divergence in the inner loop.
// ---------------------------------------------------------------------------

#define ALPHA    0.3f
#define B_ROWS   8
#define V_DIM    128000
#define R_DIM    512
#define KCHUNKS  100
#define KC       (V_DIM / KCHUNKS)       // 1280, multiple of 4

typedef __attribute__((ext_vector_type(2))) float v2f;
typedef __attribute__((ext_vector_type(8))) float v8f;

// workspace layout (in floats), total ~5.8 MB
#define WS_YPART   0                                       // KCHUNKS*8*512 = 409600
#define WS_YCOMB   (WS_YPART + KCHUNKS * B_ROWS * R_DIM)   // 8*512 = 4096
#define WS_M0      (WS_YCOMB + B_ROWS * R_DIM)             // 8*V = 1024000
#define WS_PART0   (WS_M0 + B_ROWS * V_DIM)                // 256
#define WS_PART1   (WS_PART0 + 256)                        // 256
#define WS_PART2   (WS_PART1 + 256)                        // 256
#define WS_SCAL    (WS_PART2 + 256)                        // 2

// --------------------------- reductions ------------------------------------

__global__ void k_sumsq(const float* __restrict__ x, float* __restrict__ part, int n) {
    __shared__ float s[256];
    float acc = 0.f;
    for (int i = blockIdx.x * 256 + threadIdx.x; i < n; i += gridDim.x * 256) {
        float v = x[i];
        acc += v * v;
    }
    s[threadIdx.x] = acc;
    __syncthreads();
    for (int off = 128; off > 0; off >>= 1) {
        if (threadIdx.x < (unsigned)off) s[threadIdx.x] += s[threadIdx.x + off];
        __syncthreads();
    }
    if (threadIdx.x == 0) part[blockIdx.x] = s[0];
}

__global__ void k_sumsq_dot(const float* __restrict__ m0, const float* __restrict__ lg,
                            float* __restrict__ p1, float* __restrict__ p2, int n) {
    __shared__ float s1[256];
    __shared__ float s2[256];
    float a1 = 0.f, a2 = 0.f;
    for (int i = blockIdx.x * 256 + threadIdx.x; i < n; i += gridDim.x * 256) {
        float mv = m0[i];
        float lv = lg[i];
        a1 += mv * mv;
        a2 += mv * lv;
    }
    s1[threadIdx.x] = a1;
    s2[threadIdx.x] = a2;
    __syncthreads();
    for (int off = 128; off > 0; off >>= 1) {
        if (threadIdx.x < (unsigned)off) {
            s1[threadIdx.x] += s1[threadIdx.x + off];
            s2[threadIdx.x] += s2[threadIdx.x + off];
        }
        __syncthreads();
    }
    if (threadIdx.x == 0) { p1[blockIdx.x] = s1[0]; p2[blockIdx.x] = s2[0]; }
}

__global__ void k_scalars(const float* __restrict__ p0, const float* __restrict__ p1,
                          const float* __restrict__ p2, float* __restrict__ scal) {
    __shared__ float s0[256];
    __shared__ float s1[256];
    __shared__ float s2[256];
    const int t = threadIdx.x;
    s0[t] = p0[t]; s1[t] = p1[t]; s2[t] = p2[t];
    __syncthreads();
    for (int off = 128; off > 0; off >>= 1) {
        if (t < off) { s0[t] += s0[t + off]; s1[t] += s1[t + off]; s2[t] += s2[t + off]; }
        __syncthreads();
    }
    if (t == 0) {
        float ln2 = s0[0], mn2 = s1[0], dot = s2[0];
        float l_n = sqrtf(ln2);
        float mn  = sqrtf(mn2);
        float u   = (1.f - ALPHA) / mn;     // coefficient on M0 inside z
        float tt  = ALPHA / l_n;            // coefficient on logits inside z
        // ||z||^2 = u^2*mn2 + tt^2*ln2 + 2*u*tt*<M0,logits>
        float z2 = u * u * mn2 + tt * tt * ln2 + 2.f * u * tt * dot;
        float scale = l_n / sqrtf(z2);
        scal[0] = scale * u;
        scal[1] = scale * tt;
    }
}

// --------------------------- GEMM1: Y = logits @ W (split-K) ---------------
// One wave per block. blockIdx.x in [0,8): 64 columns. blockIdx.y: K-chunk.
// A layout (f32 16x4): lane row m = lane&15, VGPR pair = K 2*(lane>>4)+{0,1}
// B layout (f32 4x16): lane col n = lane&15, VGPR pair = K 2*(lane>>4)+{0,1}
// D layout: lane col n = lane&15, VGPR g = row g + 8*(lane>>4)

__global__ __launch_bounds__(32) void k_gemm1(const float* __restrict__ logits,
                                              const float* __restrict__ W,
                                              float* __restrict__ ypart) {
    const int lane = threadIdx.x;
    const int n    = lane & 15;
    const int kb   = (lane >> 4) * 2;
    const int row8 = lane & 7;              // duplicated rows for m >= 8
    const int colBase = blockIdx.x * 64;
    const int chunk   = blockIdx.y;
    const int k0      = chunk * KC;

    const float* pA = logits + (size_t)row8 * V_DIM + k0 + kb;
    const float* pB = W + (size_t)(k0 + kb) * R_DIM + colBase + n;

    v8f acc[4] = {};

    for (int k = 0; k < KC; k += 4) {
        v2f a;
        a.x = pA[0];
        a.y = pA[1];
        __builtin_prefetch(pB + 8 * R_DIM, 0, 1);
#pragma unroll
        for (int t = 0; t < 4; ++t) {
            v2f b;
            b.x = pB[t * 16];
            b.y = pB[t * 16 + R_DIM];
            acc[t] = __builtin_amdgcn_wmma_f32_16x16x4_f32(
                false, a, false, b, (short)0, acc[t], false, false);
        }
        pA += 4;
        pB += 4 * R_DIM;
    }

    // only rows 0..7 (held by lanes 0..15) are meaningful
    if (lane < 16) {
        float* dst = ypart + (size_t)chunk * B_ROWS * R_DIM + colBase + n;
#pragma unroll
        for (int t = 0; t < 4; ++t)
#pragma unroll
            for (int g = 0; g < 8; ++g)
                dst[g * R_DIM + t * 16] = acc[t][g];
    }
}

__global__ void k_comb(const float* __restrict__ ypart, float* __restrict__ y) {
    const int i = blockIdx.x * 256 + threadIdx.x;   // 8*512 = 4096 elements
    if (i < B_ROWS * R_DIM) {
        float s = 0.f;
        for (int c = 0; c < KCHUNKS; ++c) s += ypart[(size_t)c * B_ROWS * R_DIM + i];
        y[i] = s;
    }
}

// --------------------------- GEMM2: M0 = Y @ W^T ----------------------------
// One wave per block; each wave produces 64 output columns (v values).
// B[kk][n] = W[vBase + t*16 + n][k + kk] -> contiguous along K => b64 loads.

__global__ __launch_bounds__(32) void k_gemm2(const float* __restrict__ y,
                                              const float* __restrict__ W,
                                              float* __restrict__ m0) {
    const int lane  = threadIdx.x;
    const int n     = lane & 15;
    const int kb    = (lane >> 4) * 2;
    const int row8  = lane & 7;
    const int vBase = blockIdx.x * 64;

    const float* pA = y + (size_t)row8 * R_DIM + kb;
    const float* pB = W + (size_t)(vBase + n) * R_DIM + kb;

    v8f acc[4] = {};

    for (int k = 0; k < R_DIM; k += 4) {
        v2f a;
        a.x = pA[0];
        a.y = pA[1];
        __builtin_prefetch(pB + 16, 0, 1);
#pragma unroll
        for (int t = 0; t < 4; ++t) {
            v2f b;
            b.x = pB[(size_t)t * 16 * R_DIM];
            b.y = pB[(size_t)t * 16 * R_DIM + 1];
            acc[t] = __builtin_amdgcn_wmma_f32_16x16x4_f32(
                false, a, false, b, (short)0, acc[t], false, false);
        }
        pA += 4;
        pB += 4;
    }

    if (lane < 16) {
        float* dst = m0 + vBase + n;
#pragma unroll
        for (int t = 0; t < 4; ++t)
#pragma unroll
            for (int g = 0; g < 8; ++g)
                dst[(size_t)g * V_DIM + t * 16] = acc[t][g];
    }
}

// --------------------------- final lerp + rescale ---------------------------

__global__ void k_final(const float* __restrict__ m0, const float* __restrict__ lg,
                        const float* __restrict__ scal, float* __restrict__ out, int n) {
    const float c1 = scal[0];
    const float c2 = scal[1];
    const int i = blockIdx.x * 256 + threadIdx.x;
    if (i < n) out[i] = c1 * m0[i] + c2 * lg[i];
}

// --------------------------- launcher ---------------------------------------

extern "C" void kernel_launch(void* const* d_in, const int* in_sizes, int n_in,
                              void* d_out, int out_size, void* d_ws, size_t ws_size,
                              hipStream_t stream) {
    (void)in_sizes; (void)n_in; (void)out_size; (void)ws_size;
    const float* logits = (const float*)d_in[0];   // [8, 128000]
    const float* W      = (const float*)d_in[1];   // [128000, 512]
    float*       out    = (float*)d_out;           // [8, 128000]
    float*       ws     = (float*)d_ws;            // needs ~5.8 MB

    float* ypart = ws + WS_YPART;
    float* ycomb = ws + WS_YCOMB;
    float* m0    = ws + WS_M0;
    float* p0    = ws + WS_PART0;
    float* p1    = ws + WS_PART1;
    float* p2    = ws + WS_PART2;
    float* scal  = ws + WS_SCAL;

    const int N = B_ROWS * V_DIM;                  // 1,024,000

    // 1) ||logits||^2 partials
    k_sumsq<<<256, 256, 0, stream>>>(logits, p0, N);

    // 2) Y = logits @ W  (split-K WMMA)
    k_gemm1<<<dim3(R_DIM / 64, KCHUNKS), 32, 0, stream>>>(logits, W, ypart);
    k_comb<<<(B_ROWS * R_DIM + 255) / 256, 256, 0, stream>>>(ypart, ycomb);

    // 3) M0 = Y @ W^T (WMMA)
    k_gemm2<<<V_DIM / 64, 32, 0, stream>>>(ycomb, W, m0);

    // 4) ||M0||^2 and <M0, logits> partials, then fold into c1/c2
    k_sumsq_dot<<<256, 256, 0, stream>>>(m0, logits, p1, p2, N);
    k_scalars<<<1, 256, 0, stream>>>(p0, p1, p2, scal);

    // 5) out = c1*M0 + c2*logits
    k_final<<<(N + 255) / 256, 256, 0, stream>>>(m0, logits, scal, out, N);
}